// NodeModule_4432406249915
// MI455X (gfx1250) — compile-verified
//
#include <hip/hip_runtime.h>
#include <hip/hip_bf16.h>

typedef __attribute__((ext_vector_type(2)))  float    v2f;
typedef __attribute__((ext_vector_type(8)))  float    v8f;

// Problem constants (DEG=4, V=4, C=5, K=9, N_IN=16, T_IN=64)
#define NTERMS   1025            // 16*64 + 1 (const row)
#define TTQ      1050625         // NTERMS*NTERMS
#define NCOLS    9225            // NTERMS*9
#define QELEMS   37859436        // (TTQ + NTERMS + 1) * 4 * 9
#define OUT_B0   10
#define OUT_B1   41008           // under + over + deg + deg
#define OUT_B2   37879944        // under + over_q + deg + 2deg
#define OUT_B3   37859448        // zeros(4) + over_q + zeros(4) + 2deg

// workspace layout (floats): [0]=l [1]=u [2]=a [3]=c0 | under | over | Q
#define WS_UNDER 16
#define WS_OVER  20516           // 16 + 1025*20
#define WS_Q     41016           // 20516 + 1025*20  (Q: 1025*4*9*5 = 184500 floats)

__constant__ float C4C[5] = {1.f, 4.f, 6.f, 4.f, 1.f};
__constant__ float C8C[9] = {1.f, 8.f, 28.f, 56.f, 70.f, 56.f, 28.f, 8.f, 1.f};

// ---------------- WMMA: D = A(16x4 f32) * B(4x16 f32) + C, fp32 (codegen-confirmed) ----------------
__device__ __forceinline__ v8f wmma_f32_16x16x4(v2f a, v2f b, v8f c) {
#if defined(__gfx1250__)
  return __builtin_amdgcn_wmma_f32_16x16x4_f32(false, a, false, b, (short)0, c,
                                               false, false);
#else
  (void)a; (void)b;
  return c;  // host pass dummy
#endif
}

// ---------------- small helpers ----------------
__global__ void k_init(float* ws) {
  if (threadIdx.x < 16) ws[threadIdx.x] = 0.0f;
}

__global__ void k_fill(float* dst, int n, float val) {
  int i = blockIdx.x * blockDim.x + threadIdx.x;
  if (i < n) dst[i] = val;
}

__global__ void k_copy(float* __restrict__ dst, const float* __restrict__ src, int n) {
  int i = blockIdx.x * blockDim.x + threadIdx.x;
  if (i < n) dst[i] = src[i];
}

// ---------------- build under/over term arrays (1025 x 4 x 5) ----------------
__device__ __forceinline__ int nth_nonzero(const float* w, int j) {
  int cnt = 0;
  for (int i = 0; i < 16; ++i) {
    if (w[i] != 0.0f) { if (cnt == j) return i; ++cnt; }
  }
  return 0;
}

__global__ void k_build(const float* __restrict__ ui, const float* __restrict__ oi,
                        const float* __restrict__ pw, const float* __restrict__ nw,
                        const float* __restrict__ bias, float* __restrict__ ws) {
  int idx = blockIdx.x * blockDim.x + threadIdx.x;
  if (idx >= NTERMS * 20) return;
  int t = idx / 20, rem = idx % 20, v = rem / 5, c = rem % 5;
  float uval, oval;
  if (t == NTERMS - 1) {           // appended const row
    float b = bias[0];
    uval = oval = (v == 0) ? b : 1.0f;
  } else {
    int j = t >> 6, tt = t & 63;
    int Np = 0, Nn = 0;
    for (int i = 0; i < 16; ++i) {
      if (pw[i] != 0.0f) ++Np;
      if (nw[i] != 0.0f) ++Nn;
    }
    // under = [ over_in[neg]*neg_w ; under_in[pos]*pos_w ]
    {
      const float* src; int si; float sc;
      if (j < Nn) { si = nth_nonzero(nw, j);      src = oi; sc = nw[si]; }
      else        { si = nth_nonzero(pw, j - Nn); src = ui; sc = pw[si]; }
      float val = src[((si * 64 + tt) * 4 + v) * 5 + c];
      uval = (v == 0) ? val * sc : val;
    }
    // over  = [ over_in[pos]*pos_w ; under_in[neg]*neg_w ]
    {
      const float* src; int si; float sc;
      if (j < Np) { si = nth_nonzero(pw, j);      src = oi; sc = pw[si]; }
      else        { si = nth_nonzero(nw, j - Np); src = ui; sc = nw[si]; }
      float val = src[((si * 64 + tt) * 4 + v) * 5 + c];
      oval = (v == 0) ? val * sc : val;
    }
  }
  ws[WS_UNDER + idx] = uval;
  ws[WS_OVER + idx] = oval;
}

// ---------------- interval min/max reduction: l = Σ tlo(under), u = Σ thi(over) ----------------
__device__ __forceinline__ void term_interval(const float* P, float& tlo, float& thi) {
  float lo = P[0], hi = P[0];
  for (int c = 1; c < 5; ++c) { lo = fminf(lo, P[c]); hi = fmaxf(hi, P[c]); }
  tlo = lo; thi = hi;
  for (int v = 1; v < 4; ++v) {
    float lv = P[v * 5], hv = P[v * 5];
    for (int c = 1; c < 5; ++c) {
      lv = fminf(lv, P[v * 5 + c]);
      hv = fmaxf(hv, P[v * 5 + c]);
    }
    float p1 = tlo * lv, p2 = tlo * hv, p3 = thi * lv, p4 = thi * hv;
    tlo = fminf(fminf(p1, p2), fminf(p3, p4));
    thi = fmaxf(fmaxf(p1, p2), fmaxf(p3, p4));
  }
}

__global__ void k_minmax(float* ws) {
  int t = blockIdx.x * blockDim.x + threadIdx.x;
  if (t >= NTERMS) return;
  float tlo, thi, olo, ohi;
  term_interval(ws + WS_UNDER + t * 20, tlo, thi);
  term_interval(ws + WS_OVER + t * 20, olo, ohi);
  atomicAdd(&ws[0], tlo);   // l contribution
  atomicAdd(&ws[1], ohi);   // u contribution
}

__global__ void k_finalize(float* ws) {
  float l = ws[0];
  ws[2] = 1.0f / (-4.0f * l);   // a
  ws[3] = -0.25f * l;           // c
}

// ---------------- Q[t][v][k][i] = over[t,v,k-i] * M[i,k-i,k]  (i-contiguous for B gather) ----------------
__global__ void k_Q(float* __restrict__ ws) {
  int idx = blockIdx.x * blockDim.x + threadIdx.x;
  if (idx >= NTERMS * 180) return;
  int t = idx / 180, rem = idx % 180;
  int v = rem / 45, r2 = rem % 45, k = r2 / 5, i = r2 % 5;
  int j = k - i;
  float val = 0.0f;
  if (j >= 0 && j <= 4)
    val = ws[WS_OVER + t * 20 + v * 5 + j] * C4C[i] * C4C[j] / C8C[k];
  ws[WS_Q + idx] = val;
}

// ---------------- sq[a,b,v,k] via fp32 WMMA: M=term a, N=(b,k), K=5 padded to 8 ----------------
#define MT 65     // ceil(1025/16); only tile 64 is partial (1 valid row)
#define NTL 577   // ceil(9225/16); only tile 576 is partial (9 valid cols)
#define NWAVES (4 * MT * NTL)

__global__ void __launch_bounds__(256) k_sq(const float* __restrict__ ws,
                                            float* __restrict__ qout) {
  int wave = blockIdx.x * (blockDim.x >> 5) + (threadIdx.x >> 5);
  if (wave >= NWAVES) return;              // wave-uniform: EXEC stays all-ones
  float ascale = ws[2];                    // hoisted scalar load (overlaps tile loads)
  int lane = threadIdx.x & 31;
  int v = wave & 3;
  int t2 = wave >> 2;
  int nt = t2 % NTL;
  int mt = t2 / NTL;
  int aBase = mt << 4, nBase = nt << 4;
  bool hiK = lane >= 16;
  int lane15 = lane & 15;
  bool interior = (mt < MT - 1) && (nt < NTL - 1);   // wave-uniform (~97% of tiles)

  const float* over = ws + WS_OVER;
  const float* Q    = ws + WS_Q;

  int mrow = aBase + lane15;
  int ncol = nBase + lane15;
  int bterm = ncol / 9, kk = ncol % 9;

  // A tile: 5 contiguous floats over[mrow,v,0..4]; ISA 16x4 f32 A layout
  //   (lanes 0-15 hold K0,K1; lanes 16-31 hold K2,K3; K=4 goes to the 2nd WMMA)
  // B tile: 5 contiguous floats Q[bterm,v,kk,0..4] (column n = b*9+k), same split
  v2f a0 = {0.f, 0.f}, a1 = {0.f, 0.f};
  v2f b0 = {0.f, 0.f}, b1 = {0.f, 0.f};
  if (interior) {
    const float* ap = over + mrow * 20 + v * 5;
    const float* qp = Q + ((bterm * 4 + v) * 9 + kk) * 5;
    if (!hiK) {
      a0.x = ap[0]; a0.y = ap[1]; a1.x = ap[4];
      b0.x = qp[0]; b0.y = qp[1]; b1.x = qp[4];
    } else {
      a0.x = ap[2]; a0.y = ap[3];
      b0.x = qp[2]; b0.y = qp[3];
    }
  } else {
    if (mrow < NTERMS) {
      const float* ap = over + mrow * 20 + v * 5;
      if (!hiK) { a0.x = ap[0]; a0.y = ap[1]; a1.x = ap[4]; }
      else      { a0.x = ap[2]; a0.y = ap[3]; }
    }
    if (ncol < NCOLS) {
      const float* qp = Q + ((bterm * 4 + v) * 9 + kk) * 5;
      if (!hiK) { b0.x = qp[0]; b0.y = qp[1]; b1.x = qp[4]; }
      else      { b0.x = qp[2]; b0.y = qp[3]; }
    }
  }

  v8f acc = {};
  acc = wmma_f32_16x16x4(a0, b0, acc);   // K = 0..3
  acc = wmma_f32_16x16x4(a1, b1, acc);   // K = 4 (+ zero padding)

  float sc = (v == 0) ? ascale : 1.0f;   // sq[:,0,:] *= a
  // D layout: lanes 0-15 -> M=r, lanes 16-31 -> M=r+8, N = lane&15
  int mBase = aBase + (hiK ? 8 : 0);
  if (interior) {
    float* op = qout + ((mBase * NTERMS + bterm) * 4 + v) * 9 + kk;
    const int rstride = NTERMS * 36;     // one M row in output elements
#pragma unroll
    for (int r = 0; r < 8; ++r)
      __builtin_nontemporal_store(acc[r] * sc, op + r * rstride);
  } else if (ncol < NCOLS) {
#pragma unroll
    for (int r = 0; r < 8; ++r) {
      int m = mBase + r;
      if (m < NTERMS)
        __builtin_nontemporal_store(acc[r] * sc,
                                    qout + ((m * NTERMS + bterm) * 4 + v) * 9 + kk);
    }
  }
}

// ---------------- lin rows (elevated degree) + const row ----------------
__global__ void k_linconst(const float* __restrict__ ws, float* __restrict__ qout) {
  int idx = blockIdx.x * blockDim.x + threadIdx.x;
  if (idx < NTERMS * 36) {
    int t = idx / 36, rem = idx % 36, v = rem / 9, j = rem % 9;
    float val = 0.0f;
    for (int i = 0; i < 5; ++i)
      if (j >= i && j <= i + 4)
        val += ws[WS_OVER + t * 20 + v * 5 + i] * C4C[i] * C4C[j - i] / C8C[j];
    if (v == 0) val *= 0.5f;                       // b = 0.5
    __builtin_nontemporal_store(val, qout + ((TTQ + t) * 4 + v) * 9 + j);
  } else if (idx < NTERMS * 36 + 36) {
    int e = idx - NTERMS * 36, v = e / 9, j = e % 9;
    __builtin_nontemporal_store((v == 0) ? ws[3] : 1.0f,
                                qout + ((TTQ + NTERMS) * 4 + v) * 9 + j);
  }
}

// ---------------- degree tail (8 values) ----------------
__global__ void k_deg_tail(float* dst, int mode, const int* __restrict__ udeg,
                           const int* __restrict__ odeg) {
  int tid = threadIdx.x;
  if (tid >= 8) return;
  int v = tid & 3;
  int m = 0;
  for (int n = 0; n < 16; ++n) {
    int a = udeg[n * 4 + v], b = odeg[n * 4 + v];
    m = max(m, max(a, b));
  }
  float d = (float)m;
  bool hi = tid >= 4;
  float val;
  if (mode == 0)      val = d;                    // (deg, deg)
  else if (mode == 1) val = hi ? 2.f * d : d;     // (deg, 2deg)
  else                val = hi ? 2.f * d : 0.f;   // (0,   2deg)
  dst[tid] = val;
}

extern "C" void kernel_launch(void* const* d_in, const int* in_sizes, int n_in,
                              void* d_out, int out_size, void* d_ws, size_t ws_size,
                              hipStream_t stream) {
  (void)in_sizes; (void)n_in; (void)ws_size;
  const float* ui   = (const float*)d_in[0];
  const float* oi   = (const float*)d_in[1];
  const float* pw   = (const float*)d_in[2];
  const float* nw   = (const float*)d_in[3];
  const float* bias = (const float*)d_in[4];
  const int*   udeg = (const int*)d_in[5];
  const int*   odeg = (const int*)d_in[6];
  float* out = (float*)d_out;
  float* ws  = (float*)d_ws;

  k_init<<<1, 32, 0, stream>>>(ws);
  k_build<<<(NTERMS * 20 + 255) / 256, 256, 0, stream>>>(ui, oi, pw, nw, bias, ws);
  k_minmax<<<(NTERMS + 255) / 256, 256, 0, stream>>>(ws);
  k_finalize<<<1, 1, 0, stream>>>(ws);

  if (out_size == OUT_B0) {
    k_fill<<<1, 32, 0, stream>>>(out, OUT_B0, 0.0f);
  } else if (out_size == OUT_B1) {
    k_copy<<<(NTERMS * 20 + 255) / 256, 256, 0, stream>>>(out, ws + WS_UNDER, NTERMS * 20);
    k_copy<<<(NTERMS * 20 + 255) / 256, 256, 0, stream>>>(out + NTERMS * 20, ws + WS_OVER, NTERMS * 20);
    k_deg_tail<<<1, 8, 0, stream>>>(out + 2 * NTERMS * 20, 0, udeg, odeg);
  } else {
    int qbase, mode;
    if (out_size == OUT_B2) {
      qbase = NTERMS * 20;  mode = 1;
      k_copy<<<(NTERMS * 20 + 255) / 256, 256, 0, stream>>>(out, ws + WS_UNDER, NTERMS * 20);
    } else {                 // OUT_B3 (or unknown: treat as B3 layout)
      qbase = 4;  mode = 2;
      k_fill<<<1, 32, 0, stream>>>(out, 4, 0.0f);
    }
    k_Q<<<(NTERMS * 180 + 255) / 256, 256, 0, stream>>>(ws);
    k_sq<<<(NWAVES + 7) / 8, 256, 0, stream>>>(ws, out + qbase);
    k_linconst<<<(NTERMS * 36 + 36 + 255) / 256, 256, 0, stream>>>(ws, out + qbase);
    k_deg_tail<<<1, 8, 0, stream>>>(out + qbase + QELEMS, mode, udeg, odeg);
  }
}